// SDETM_46248207844103
// MI455X (gfx1250) — compile-verified
//
#include <hip/hip_runtime.h>

#define K_   50
#define T_   50
#define RHO_ 300
#define V_   20000
#define TH_  800
#define EH_  200
#define S_   10
#define B_   256
#define ESP_ 52          // padded eta_std row stride (K_ rounded up to mult of 4)
#define DELTA_     0.005f
#define LOG_DELTA_ -5.2983173665480363f
#define EPS_       1e-6f

typedef __attribute__((ext_vector_type(2))) float v2f;
typedef __attribute__((ext_vector_type(8))) float v8f;
typedef __attribute__((ext_vector_type(4))) int   v4i;

#if defined(__gfx1250__) && __has_builtin(__builtin_amdgcn_global_load_async_to_lds_b128)
#define ASYNC_LDS 1
typedef __attribute__((address_space(1))) v4i gv4i;   // global-AS int4
typedef __attribute__((address_space(3))) v4i lv4i;   // LDS-AS int4
#endif

__device__ __forceinline__ v8f wmma_f32_4(v2f a, v2f b, v8f c) {
  // V_WMMA_F32_16X16X4_F32 : D = A(16x4) * B(4x16) + C(16x16), all fp32
  return __builtin_amdgcn_wmma_f32_16x16x4_f32(false, a, false, b, (short)0, c, false, false);
}

__device__ __forceinline__ float sigf(float x) { return 1.0f / (1.0f + expf(-x)); }

__device__ __forceinline__ void online_merge(float& m, float& s, float mo, float so) {
  float mn = fmaxf(m, mo);
  s = s * expf(m - mn) + so * expf(mo - mn);
  m = mn;
}

// ---------------- accumulator zeroing ----------------
__global__ void k_zero_acc(float* acc) {
  if (threadIdx.x < 8) acc[threadIdx.x] = 0.0f;
}

// ---------------- kl_alpha ----------------
__global__ void k_kl_alpha(const float* __restrict__ mu, const float* __restrict__ ls,
                           float* acc) {
  __shared__ float red[256];
  int idx = blockIdx.x * blockDim.x + threadIdx.x;
  float v = 0.0f;
  if (idx < K_ * T_ * RHO_) {
    int k   = idx / (T_ * RHO_);
    int rem = idx - k * (T_ * RHO_);
    int t   = rem / RHO_;
    float m = mu[idx], l = ls[idx];
    if (t == 0) {
      v = 0.5f * ((expf(l) + m * m) / (1.0f + EPS_) - 1.0f - l);
    } else {
      float d = m - mu[idx - RHO_];
      v = 0.5f * ((expf(l) + d * d) / (DELTA_ + EPS_) - 1.0f + LOG_DELTA_ - l);
    }
  }
  red[threadIdx.x] = v; __syncthreads();
  for (int off = 128; off > 0; off >>= 1) {
    if (threadIdx.x < off) red[threadIdx.x] += red[threadIdx.x + off];
    __syncthreads();
  }
  if (threadIdx.x == 0) atomicAdd(&acc[0], red[0]);
}

// ======== WMMA GEMM: C[MxN] = A[MxKd] . B[NxKd]^T + bias, Kd % 4 == 0 ========
// One wave computes a 32(M) x 16(N) tile: two accumulators share each B fragment.
__global__ void k_gemm_wmma(const float* __restrict__ A, const float* __restrict__ B,
                            const float* __restrict__ bias, float* __restrict__ C,
                            int M, int N, int Kd, int act) {
  int lane = threadIdx.x & 31;
  int hlf = lane >> 4, l16 = lane & 15;
  int m0 = blockIdx.y * 32, n0 = blockIdx.x * 16;
  int ra0 = m0 + l16;      if (ra0 >= M) ra0 = M - 1;
  int ra1 = m0 + 16 + l16; if (ra1 >= M) ra1 = M - 1;
  int cb  = n0 + l16;      if (cb  >= N) cb  = N - 1;
  const float* arow0 = A + (size_t)ra0 * (size_t)Kd;
  const float* arow1 = A + (size_t)ra1 * (size_t)Kd;
  const float* brow  = B + (size_t)cb  * (size_t)Kd;
  v8f c0 = {}, c1 = {};
  #pragma unroll 4
  for (int k = 0; k < Kd; k += 4) {
    if ((k & 255) == 0) __builtin_prefetch(brow + k + 256, 0, 1);
    v2f a0 = *(const v2f*)(arow0 + k + 2 * hlf);
    v2f a1 = *(const v2f*)(arow1 + k + 2 * hlf);
    v2f b;
    b.x = brow[k + hlf];
    b.y = brow[k + 2 + hlf];
    c0 = wmma_f32_4(a0, b, c0);
    c1 = wmma_f32_4(a1, b, c1);
  }
  int col = n0 + l16;
  float bv = bias[cb];
  #pragma unroll
  for (int i = 0; i < 8; ++i) {
    int r0 = m0 + i + 8 * hlf;
    int r1 = r0 + 16;
    if (col < N) {
      if (r0 < M) {
        float v = c0[i] + bv;
        if (act) v = fmaxf(v, 0.0f);
        C[(size_t)r0 * N + col] = v;
      }
      if (r1 < M) {
        float v = c1[i] + bv;
        if (act) v = fmaxf(v, 0.0f);
        C[(size_t)r1 * N + col] = v;
      }
    }
  }
}

// ======== WMMA GEMM with concatenated A = [A1 (K1) | A2 (K2, zero-padded to K2pad)] ====
// B rows have stride Kd = K1 + K2. K1 % 4 == 0, K2pad % 4 == 0.
__global__ void k_gemm_wmma_cat(const float* __restrict__ A1, int K1,
                                const float* __restrict__ A2, int K2pad, int K2,
                                const float* __restrict__ B, int Kd,
                                const float* __restrict__ bias, float* __restrict__ C,
                                int M, int N, int act) {
  int lane = threadIdx.x & 31;
  int hlf = lane >> 4, l16 = lane & 15;
  int m0 = blockIdx.y * 32, n0 = blockIdx.x * 16;
  int ra0 = m0 + l16;      if (ra0 >= M) ra0 = M - 1;
  int ra1 = m0 + 16 + l16; if (ra1 >= M) ra1 = M - 1;
  int cb  = n0 + l16;      if (cb  >= N) cb  = N - 1;
  const float* a1r0 = A1 + (size_t)ra0 * (size_t)K1;
  const float* a1r1 = A1 + (size_t)ra1 * (size_t)K1;
  const float* a2r0 = A2 + (size_t)ra0 * (size_t)K2pad;
  const float* a2r1 = A2 + (size_t)ra1 * (size_t)K2pad;
  const float* brow = B + (size_t)cb * (size_t)Kd;
  v8f c0 = {}, c1 = {};
  #pragma unroll 4
  for (int k = 0; k < K1; k += 4) {          // unguarded main loop
    if ((k & 255) == 0) __builtin_prefetch(brow + k + 256, 0, 1);
    v2f a0 = *(const v2f*)(a1r0 + k + 2 * hlf);
    v2f a1 = *(const v2f*)(a1r1 + k + 2 * hlf);
    v2f b;
    b.x = brow[k + hlf];
    b.y = brow[k + 2 + hlf];
    c0 = wmma_f32_4(a0, b, c0);
    c1 = wmma_f32_4(a1, b, c1);
  }
  for (int k2 = 0; k2 < K2pad; k2 += 4) {    // short tail: A2 zero-padded, B clamp-selected
    v2f a0 = *(const v2f*)(a2r0 + k2 + 2 * hlf);
    v2f a1 = *(const v2f*)(a2r1 + k2 + 2 * hlf);
    int kb0 = K1 + k2 + hlf, kb1 = K1 + k2 + 2 + hlf;
    float bx = brow[kb0 < Kd ? kb0 : Kd - 1];
    float by = brow[kb1 < Kd ? kb1 : Kd - 1];
    v2f b;
    b.x = (kb0 < Kd) ? bx : 0.0f;            // pairs with A2 pad zeros anyway
    b.y = (kb1 < Kd) ? by : 0.0f;
    c0 = wmma_f32_4(a0, b, c0);
    c1 = wmma_f32_4(a1, b, c1);
  }
  int col = n0 + l16;
  float bv = bias[cb];
  #pragma unroll
  for (int i = 0; i < 8; ++i) {
    int r0 = m0 + i + 8 * hlf;
    int r1 = r0 + 16;
    if (col < N) {
      if (r0 < M) {
        float v = c0[i] + bv;
        if (act) v = fmaxf(v, 0.0f);
        C[(size_t)r0 * N + col] = v;
      }
      if (r1 < M) {
        float v = c1[i] + bv;
        if (act) v = fmaxf(v, 0.0f);
        C[(size_t)r1 * N + col] = v;
      }
    }
  }
}

// ---------------- LSTM over T (sequential), state in LDS ----------------
__global__ void k_lstm(const float* __restrict__ mapped,
                       const float* __restrict__ Wih, const float* __restrict__ Whh,
                       const float* __restrict__ bih, const float* __restrict__ bhh,
                       float* __restrict__ lstm_out) {
  __shared__ float hS[S_ * EH_], cS[S_ * EH_], zS[S_ * 4 * EH_];
  int tid = threadIdx.x;
  for (int i = tid; i < S_ * EH_; i += blockDim.x) { hS[i] = 0.0f; cS[i] = 0.0f; }
  __syncthreads();
  for (int t = 0; t < T_; ++t) {
    for (int gi = tid; gi < S_ * 4 * EH_; gi += blockDim.x) {
      int s = gi / (4 * EH_), g = gi - s * (4 * EH_);
      const float* x  = mapped + ((size_t)s * T_ + t) * EH_;
      const float* hh = hS + s * EH_;
      const float* wi = Wih + (size_t)g * EH_;
      const float* wh = Whh + (size_t)g * EH_;
      float z = bih[g] + bhh[g];
      for (int e = 0; e < EH_; ++e) z += x[e] * wi[e] + hh[e] * wh[e];
      zS[gi] = z;
    }
    __syncthreads();
    for (int ei = tid; ei < S_ * EH_; ei += blockDim.x) {
      int s = ei / EH_, e = ei - s * EH_;
      const float* zr = zS + s * 4 * EH_;
      float ig = sigf(zr[e]);
      float fg = sigf(zr[EH_ + e]);
      float gg = tanhf(zr[2 * EH_ + e]);
      float og = sigf(zr[3 * EH_ + e]);
      float cn = fg * cS[ei] + ig * gg;
      float hn = og * tanhf(cn);
      cS[ei] = cn; hS[ei] = hn;
      lstm_out[((size_t)t * S_ + s) * EH_ + e] = hn;
    }
    __syncthreads();
  }
}

// ---------------- eta recurrence + kl_eta ----------------
__global__ void k_eta(const float* __restrict__ lstm_out,
                      const float* __restrict__ muW, const float* __restrict__ mub,
                      const float* __restrict__ lsW, const float* __restrict__ lsb,
                      float* __restrict__ etas, float* acc) {
  __shared__ float prev[S_ * K_], tmp[S_ * K_], red[512];
  int tid = threadIdx.x;
  if (tid < S_ * K_) prev[tid] = 0.0f;
  __syncthreads();
  float kl = 0.0f;
  const int IN = EH_ + K_;
  for (int t = 0; t < T_; ++t) {
    if (tid < S_ * K_) {
      int s = tid / K_, k = tid - s * K_;
      const float* lo = lstm_out + ((size_t)t * S_ + s) * EH_;
      const float* wm = muW + (size_t)k * IN;
      const float* wl = lsW + (size_t)k * IN;
      float muv = mub[k], lsv = lsb[k];
      for (int j = 0; j < EH_; ++j) { float x = lo[j]; muv += x * wm[j]; lsv += x * wl[j]; }
      if (t > 0) {
        const float* ep = prev + s * K_;
        for (int j = 0; j < K_; ++j) { float x = ep[j]; muv += x * wm[EH_ + j]; lsv += x * wl[EH_ + j]; }
        lsv = fminf(fmaxf(lsv, -10.0f), 10.0f);
        float d = muv - prev[tid];
        kl += 0.5f * ((expf(lsv) + d * d) / (DELTA_ + EPS_) - 1.0f + LOG_DELTA_ - lsv);
      } else {
        kl += 0.5f * ((expf(lsv) + muv * muv) / (1.0f + EPS_) - 1.0f - lsv);
      }
      tmp[tid] = muv;
      etas[((size_t)s * T_ + t) * K_ + k] = muv;
    }
    __syncthreads();
    if (tid < S_ * K_) prev[tid] = tmp[tid];
    __syncthreads();
  }
  red[tid] = kl; __syncthreads();
  for (int off = 256; off > 0; off >>= 1) {
    if (tid < off) red[tid] += red[tid + off];
    __syncthreads();
  }
  if (tid == 0) atomicAdd(&acc[1], red[0]);
}

// ---------------- gather eta_std[b,k] = etas[sources[b], times[b], k]; pad to ESP_ ----
__global__ void k_gather_eta(const float* __restrict__ etas, const int* __restrict__ sources,
                             const int* __restrict__ times, float* __restrict__ eta_std) {
  int idx = blockIdx.x * blockDim.x + threadIdx.x;
  if (idx < B_ * ESP_) {
    int b = idx / ESP_, k = idx - b * ESP_;
    eta_std[idx] = (k < K_) ? etas[((size_t)sources[b] * T_ + times[b]) * K_ + k] : 0.0f;
  }
}

// ---------------- theta softmax + kl_theta + pred_loss ----------------
__global__ void k_theta(const float* __restrict__ mu_t, const float* __restrict__ ls_t,
                        const float* __restrict__ eta_std,
                        const float* __restrict__ clsW, const float* __restrict__ clsb,
                        const int* __restrict__ sources,
                        float* __restrict__ theta, float* acc) {
  __shared__ float red[64], thS[K_], lgS[16];
  int b = blockIdx.x, tid = threadIdx.x;
  float muv = (tid < K_) ? mu_t[(size_t)b * K_ + tid] : -3.0e38f;
  red[tid] = muv; __syncthreads();
  for (int off = 32; off > 0; off >>= 1) { if (tid < off) red[tid] = fmaxf(red[tid], red[tid + off]); __syncthreads(); }
  float mx = red[0]; __syncthreads();
  float ex = (tid < K_) ? expf(muv - mx) : 0.0f;
  red[tid] = ex; __syncthreads();
  for (int off = 32; off > 0; off >>= 1) { if (tid < off) red[tid] += red[tid + off]; __syncthreads(); }
  float denom = red[0]; __syncthreads();
  float th = ex / denom;
  if (tid < K_) { thS[tid] = th; theta[(size_t)b * K_ + tid] = th; }
  float klv = 0.0f;
  if (tid < K_) {
    float ls = ls_t[(size_t)b * K_ + tid];
    ls = fminf(fmaxf(ls, -10.0f), 10.0f);
    float d = muv - eta_std[(size_t)b * ESP_ + tid];
    klv = 0.5f * ((expf(ls) + d * d) / (1.0f + EPS_) - 1.0f - ls);
  }
  red[tid] = klv; __syncthreads();
  for (int off = 32; off > 0; off >>= 1) { if (tid < off) red[tid] += red[tid + off]; __syncthreads(); }
  if (tid == 0) atomicAdd(&acc[2], red[0]);
  __syncthreads();
  if (tid < S_) {
    float lg = clsb[tid];
    const float* w = clsW + (size_t)tid * K_;
    for (int k = 0; k < K_; ++k) lg += thS[k] * w[k];
    lgS[tid] = lg;
  }
  __syncthreads();
  if (tid == 0) {
    float m = lgS[0];
    for (int s = 1; s < S_; ++s) m = fmaxf(m, lgS[s]);
    float se = 0.0f;
    for (int s = 0; s < S_; ++s) se += expf(lgS[s] - m);
    atomicAdd(&acc[4], m + logf(se) - lgS[sources[b]]);
  }
}

// ---------------- pass 1: per (t,k) row max / sumexp over V (online softmax) --------
__global__ void k_beta_norms(const float* __restrict__ muA, const float* __restrict__ rhoW,
                             float* __restrict__ Mrow, float* __restrict__ Zrow) {
  __shared__ float lm[8][16], lz[8][16];
  int r0 = blockIdx.x * 16;
  int wid = threadIdx.x >> 5;
  int lane = threadIdx.x & 31;
  int hlf = lane >> 4, l16 = lane & 15;
  int row = r0 + l16; if (row >= T_ * K_) row = 0;
  int tt = row / K_, kk = row - tt * K_;
  const float* arow = muA + (size_t)kk * T_ * RHO_ + (size_t)tt * RHO_; // alphas[t,k,:]
  float m[8], z[8];
  #pragma unroll
  for (int i = 0; i < 8; ++i) { m[i] = -3.0e38f; z[i] = 0.0f; }
  for (int j = wid; j < V_ / 16; j += 8) {
    const float* brow = rhoW + (size_t)(j * 16 + l16) * RHO_;
    v8f c = {};
    #pragma unroll 5
    for (int k = 0; k < RHO_; k += 4) {
      v2f a = *(const v2f*)(arow + k + 2 * hlf);
      v2f b;
      b.x = brow[k + hlf];
      b.y = brow[k + 2 + hlf];
      c = wmma_f32_4(a, b, c);
    }
    #pragma unroll
    for (int i = 0; i < 8; ++i) {
      float v = c[i];
      float mn = fmaxf(m[i], v);
      z[i] = z[i] * expf(m[i] - mn) + expf(v - mn);
      m[i] = mn;
    }
  }
  #pragma unroll
  for (int i = 0; i < 8; ++i) {
    for (int msk = 8; msk >= 1; msk >>= 1) {       // reduce over the 16 N-lanes of each half
      float mo = __shfl_xor(m[i], msk, 32);
      float zo = __shfl_xor(z[i], msk, 32);
      online_merge(m[i], z[i], mo, zo);
    }
  }
  if (l16 == 0) {
    #pragma unroll
    for (int i = 0; i < 8; ++i) { int ri = i + 8 * hlf; lm[wid][ri] = m[i]; lz[wid][ri] = z[i]; }
  }
  __syncthreads();
  if (threadIdx.x < 16) {
    float M0 = lm[0][threadIdx.x], Z0 = lz[0][threadIdx.x];
    for (int w = 1; w < 8; ++w) online_merge(M0, Z0, lm[w][threadIdx.x], lz[w][threadIdx.x]);
    int rr = r0 + threadIdx.x;
    if (rr < T_ * K_) { Mrow[rr] = M0; Zrow[rr] = Z0; }
  }
}

// ---------------- pass 2: recompute logit tiles per (t, v-tile), fold into NLL -------
// rho_W rows for this v-tile are staged once into LDS (async DMA when available)
// and reused across the 4 k-tiles.
__global__ void k_nll(const float* __restrict__ muA, const float* __restrict__ rhoW,
                      const float* __restrict__ Mrow, const float* __restrict__ Zrow,
                      const float* __restrict__ theta, const float* __restrict__ bows,
                      const int* __restrict__ times, float* acc) {
  __shared__ int doclist[B_];
  __shared__ int ndocs_s;
  __shared__ float p[B_][16];
  __shared__ __align__(16) float Bs[16][RHO_];
  int t = blockIdx.y;
  int v0 = blockIdx.x * 16;
  int lane = threadIdx.x;
  int hlf = lane >> 4, l16 = lane & 15;
  if (lane == 0) {
    int n = 0;
    for (int b = 0; b < B_; ++b) if (times[b] == t) doclist[n++] = b;
    ndocs_s = n;
  }
  // stage B tile: 16 rows x 300 floats, 16B-aligned segments
  for (int idx = lane; idx < 16 * (RHO_ / 4); idx += 32) {
    int r = idx / (RHO_ / 4), seg = idx - r * (RHO_ / 4);
    const float* g = rhoW + (size_t)(v0 + r) * RHO_ + seg * 4;
    float* l = &Bs[r][seg * 4];
#if defined(ASYNC_LDS)
    __builtin_amdgcn_global_load_async_to_lds_b128(
        (gv4i*)(uintptr_t)g, (lv4i*)(uintptr_t)l, 0, 0);
#else
    *(float4*)l = *(const float4*)g;
#endif
  }
#if defined(ASYNC_LDS)
#if __has_builtin(__builtin_amdgcn_s_wait_asynccnt)
  __builtin_amdgcn_s_wait_asynccnt(0);
#else
  asm volatile("s_wait_asynccnt 0" ::: "memory");
#endif
#endif
  __syncthreads();
  int nd = ndocs_s;
  if (nd == 0) return;
  for (int d = 0; d < nd; ++d) if (lane < 16) p[d][lane] = 0.0f;
  __syncthreads();
  for (int kt = 0; kt < 4; ++kt) {                 // K=50 padded to 64 rows
    int ra = kt * 16 + l16; if (ra >= K_) ra = 0;
    const float* arow = muA + (size_t)ra * T_ * RHO_ + (size_t)t * RHO_;
    v8f c = {};
    #pragma unroll 5
    for (int k = 0; k < RHO_; k += 4) {
      v2f a = *(const v2f*)(arow + k + 2 * hlf);
      v2f b;
      b.x = Bs[l16][k + hlf];
      b.y = Bs[l16][k + 2 + hlf];
      c = wmma_f32_4(a, b, c);
    }
    float e[8]; int rix[8];
    #pragma unroll
    for (int i = 0; i < 8; ++i) {
      int r = kt * 16 + i + 8 * hlf;
      rix[i] = r;
      if (r < K_) {
        int g = t * K_ + r;
        e[i] = expf(c[i] - Mrow[g]) / Zrow[g];     // beta[t,r,v] for this lane's column
      } else {
        e[i] = 0.0f;
      }
    }
    for (int d = 0; d < nd; ++d) {
      int b = doclist[d];
      const float* thb = theta + (size_t)b * K_;
      float part = 0.0f;
      #pragma unroll
      for (int i = 0; i < 8; ++i) if (rix[i] < K_) part += thb[rix[i]] * e[i];
      part += __shfl_xor(part, 16, 32);            // combine the two row-halves
      if (lane < 16) p[d][lane] += part;
    }
  }
  __syncthreads();
  float tot = 0.0f;
  for (int d = 0; d < nd; ++d) {
    int b = doclist[d];
    float v = 0.0f;
    if (lane < 16) v = bows[(size_t)b * V_ + v0 + lane] * logf(p[d][lane]);
    for (int msk = 8; msk >= 1; msk >>= 1) v += __shfl_xor(v, msk, 32);
    tot += v;
  }
  if (lane == 0) atomicAdd(&acc[3], tot);
}

// ---------------- finalize ----------------
__global__ void k_finalize(const float* acc, const int* num_docs, float* out) {
  if (threadIdx.x == 0) {
    float coeff = (float)num_docs[0] / (float)B_;
    float kla = acc[0], kle = acc[1];
    float klt = coeff * acc[2];
    float nll = -coeff * acc[3];
    float pl  = coeff * acc[4];
    out[0] = nll + kla + kle + klt + pl;
    out[1] = nll; out[2] = kla; out[3] = kle; out[4] = klt; out[5] = pl;
  }
}

extern "C" void kernel_launch(void* const* d_in, const int* in_sizes, int n_in,
                              void* d_out, int out_size, void* d_ws, size_t ws_size,
                              hipStream_t stream) {
  const float* bows      = (const float*)d_in[1];
  const float* nbows     = (const float*)d_in[2];
  const int*   times     = (const int*)d_in[3];
  const int*   sources   = (const int*)d_in[4];
  const float* rnn_inp   = (const float*)d_in[5];
  const int*   ndocs     = (const int*)d_in[6];
  const float* mu_a      = (const float*)d_in[7];
  const float* ls_a      = (const float*)d_in[8];
  const float* rho_W     = (const float*)d_in[9];
  const float* W1        = (const float*)d_in[10];
  const float* b1        = (const float*)d_in[11];
  const float* W2        = (const float*)d_in[12];
  const float* b2        = (const float*)d_in[13];
  const float* mu_th_W   = (const float*)d_in[14];
  const float* mu_th_b   = (const float*)d_in[15];
  const float* ls_th_W   = (const float*)d_in[16];
  const float* ls_th_b   = (const float*)d_in[17];
  const float* eta_map_W = (const float*)d_in[18];
  const float* eta_map_b = (const float*)d_in[19];
  const float* Wih       = (const float*)d_in[20];
  const float* Whh       = (const float*)d_in[21];
  const float* bih       = (const float*)d_in[22];
  const float* bhh       = (const float*)d_in[23];
  const float* mu_eta_W  = (const float*)d_in[24];
  const float* mu_eta_b  = (const float*)d_in[25];
  const float* ls_eta_W  = (const float*)d_in[26];
  const float* ls_eta_b  = (const float*)d_in[27];
  const float* cls_W     = (const float*)d_in[28];
  const float* cls_b     = (const float*)d_in[29];

  float* ws = (float*)d_ws;
  size_t off = 0;
  float* acc     = ws + off; off += 16;
  float* mapped  = ws + off; off += (size_t)S_ * T_ * EH_;   // [S,T,EH]
  float* lstm_o  = ws + off; off += (size_t)T_ * S_ * EH_;   // [T,S,EH]
  float* etas    = ws + off; off += (size_t)S_ * T_ * K_;    // [S,T,K]
  float* eta_std = ws + off; off += (size_t)B_ * ESP_;       // zero-padded rows
  float* h1      = ws + off; off += (size_t)B_ * TH_;
  float* h2      = ws + off; off += (size_t)B_ * TH_;
  float* mu_t    = ws + off; off += (size_t)B_ * K_;
  float* ls_t    = ws + off; off += (size_t)B_ * K_;
  float* theta   = ws + off; off += (size_t)B_ * K_;
  float* Mrow    = ws + off; off += 2560;
  float* Zrow    = ws + off; off += 2560;

  k_zero_acc<<<1, 32, 0, stream>>>(acc);
  k_kl_alpha<<<(K_ * T_ * RHO_ + 255) / 256, 256, 0, stream>>>(mu_a, ls_a, acc);
  // mapped[500x200] = rnn_inp[500x20000] . eta_map_W[200x20000]^T + b
  k_gemm_wmma<<<dim3((EH_ + 15) / 16, (S_ * T_ + 31) / 32), 32, 0, stream>>>(
      rnn_inp, eta_map_W, eta_map_b, mapped, S_ * T_, EH_, V_, 0);
  k_lstm<<<1, 256, 0, stream>>>(mapped, Wih, Whh, bih, bhh, lstm_o);
  k_eta<<<1, 512, 0, stream>>>(lstm_o, mu_eta_W, mu_eta_b, ls_eta_W, ls_eta_b, etas, acc);
  k_gather_eta<<<(B_ * ESP_ + 255) / 256, 256, 0, stream>>>(etas, sources, times, eta_std);
  // h1[256x800] = relu([nbows | eta_std][256x20050] . W1[800x20050]^T + b1)
  k_gemm_wmma_cat<<<dim3((TH_ + 15) / 16, (B_ + 31) / 32), 32, 0, stream>>>(
      nbows, V_, eta_std, ESP_, K_, W1, V_ + K_, b1, h1, B_, TH_, 1);
  k_gemm_wmma<<<dim3((TH_ + 15) / 16, (B_ + 31) / 32), 32, 0, stream>>>(
      h1, W2, b2, h2, B_, TH_, TH_, 1);
  k_gemm_wmma<<<dim3((K_ + 15) / 16, (B_ + 31) / 32), 32, 0, stream>>>(
      h2, mu_th_W, mu_th_b, mu_t, B_, K_, TH_, 0);
  k_gemm_wmma<<<dim3((K_ + 15) / 16, (B_ + 31) / 32), 32, 0, stream>>>(
      h2, ls_th_W, ls_th_b, ls_t, B_, K_, TH_, 0);
  k_theta<<<B_, 64, 0, stream>>>(mu_t, ls_t, eta_std, cls_W, cls_b, sources, theta, acc);
  k_beta_norms<<<(T_ * K_ + 15) / 16, 256, 0, stream>>>(mu_a, rho_W, Mrow, Zrow);
  k_nll<<<dim3(V_ / 16, T_), 32, 0, stream>>>(mu_a, rho_W, Mrow, Zrow, theta, bows, times, acc);
  k_finalize<<<1, 1, 0, stream>>>(acc, ndocs, (float*)d_out);

  (void)in_sizes; (void)n_in; (void)out_size; (void)ws_size;
}